// SimpleGNN_66958540144770
// MI455X (gfx1250) — compile-verified
//
#include <hip/hip_runtime.h>
#include <hip/hip_bf16.h>

typedef __attribute__((ext_vector_type(2))) float v2f;
typedef __attribute__((ext_vector_type(8))) float v8f;

// ---------------------------------------------------------------------------
// Utility: fill a float buffer with a constant (graph-capture safe memset)
// ---------------------------------------------------------------------------
__global__ __launch_bounds__(256) void fill_f32(float* __restrict__ p, float v, size_t n) {
    size_t i = (size_t)blockIdx.x * 256 + threadIdx.x;
    if (i < n) p[i] = v;
}

// ---------------------------------------------------------------------------
// Degree accumulation over real edges (self-loop contribution pre-filled = 1)
// ---------------------------------------------------------------------------
__global__ __launch_bounds__(256) void degree_kernel(const int* __restrict__ dst,
                                                     float* __restrict__ deg, int E) {
    int e = blockIdx.x * 256 + threadIdx.x;
    if (e < E) atomicAdd(&deg[dst[e]], 1.0f);
}

// deg -> deg^{-1/2} in place
__global__ __launch_bounds__(256) void rsqrt_kernel(float* __restrict__ d, int N) {
    int i = blockIdx.x * 256 + threadIdx.x;
    if (i < N) {
        float v = d[i];
        d[i] = (v > 0.0f) ? rsqrtf(v) : 0.0f;
    }
}

// ---------------------------------------------------------------------------
// WMMA fp32 GEMM: C[M x Nout] = A[M x K] @ B[K x Nout] (+bias)(+relu)
// One wave (32 threads) computes one 16x16 tile using V_WMMA_F32_16X16X4_F32.
// Requires M % 16 == 0, K % 4 == 0 (true here: M in {50000,512}, K = 128).
// Nout need not be a multiple of 16 (O = 138): out-of-range lanes load a
// CLAMPED column (keeps inner loop branchless / EXEC untouched); stores are
// guarded. EXEC is all-ones at every WMMA as required.
// ---------------------------------------------------------------------------
__global__ __launch_bounds__(32)
void wmma_gemm_f32(const float* __restrict__ A, const float* __restrict__ B,
                   const float* __restrict__ bias, float* __restrict__ C,
                   int K, int Nout, int relu) {
    const int lane = threadIdx.x;        // 0..31
    const int half = lane >> 4;          // 0 or 1
    const int l15  = lane & 15;
    const int row0 = blockIdx.x * 16;    // M tile
    const int col  = blockIdx.y * 16 + l15;
    const int colc = (col < Nout) ? col : (Nout - 1);   // clamped, always in-bounds

    v8f acc = {};

    const float* arow = A + (size_t)(row0 + l15) * K;
    const float* bcol = B + colc;
    for (int kk = 0; kk < K; kk += 4) {
        const int kb = kk + half * 2;
        // A 16x4 tile: lanes 0-15 hold K=kk,kk+1 ; lanes 16-31 hold K=kk+2,kk+3
        v2f a;
        a.x = arow[kb];
        a.y = arow[kb + 1];
        // B 4x16 tile: row-of-B striped across lanes, symmetric K split
        v2f b;
        b.x = bcol[(size_t)kb * Nout];
        b.y = bcol[(size_t)(kb + 1) * Nout];
        acc = __builtin_amdgcn_wmma_f32_16x16x4_f32(
            /*neg_a=*/false, a, /*neg_b=*/false, b,
            /*c_mod=*/(short)0, acc, /*reuse_a=*/false, /*reuse_b=*/false);
    }

    if (col < Nout) {
        const float bv = bias ? bias[col] : 0.0f;
#pragma unroll
        for (int r = 0; r < 8; ++r) {
            const int row = row0 + r + half * 8;   // C/D layout: VGPR r -> M = r (+8 for hi half)
            float v = acc[r] + bv;
            if (relu) v = fmaxf(v, 0.0f);
            C[(size_t)row * Nout + col] = v;
        }
    }
}

// ---------------------------------------------------------------------------
// Edge scatter: acc[dst] += h[src] * dis[src]*dis[dst].  One wave per edge,
// each lane handles a float4 feature group (H == 128 -> 32 groups).
// Atomics resolve in L2; the 25.6 MB feature matrix is L2-resident.
// ---------------------------------------------------------------------------
__global__ __launch_bounds__(256)
void scatter_edges(const float* __restrict__ h, const float* __restrict__ dis,
                   const int* __restrict__ src, const int* __restrict__ dst,
                   float* __restrict__ acc, int E) {
    const int e = blockIdx.x * 8 + (threadIdx.x >> 5);
    if (e >= E) return;
    const int lane = threadIdx.x & 31;
    const int s = src[e];
    const int d = dst[e];
    const float nrm = dis[s] * dis[d];
    const float4 v = ((const float4*)(h + (size_t)s * 128))[lane];
    float* o = acc + (size_t)d * 128 + (size_t)lane * 4;
    atomicAdd(o + 0, v.x * nrm);
    atomicAdd(o + 1, v.y * nrm);
    atomicAdd(o + 2, v.z * nrm);
    atomicAdd(o + 3, v.w * nrm);
}

// ---------------------------------------------------------------------------
// Post-scatter: add self-loop contribution (h_pre * dis^2), bias, ReLU.
// ---------------------------------------------------------------------------
__global__ __launch_bounds__(256)
void finish_layer(float* __restrict__ acc, const float* __restrict__ hpre,
                  const float* __restrict__ dis, const float* __restrict__ bias,
                  size_t total) {
    size_t idx = (size_t)blockIdx.x * 256 + threadIdx.x;
    if (idx >= total) return;
    const int i = (int)(idx >> 7);     // node (H == 128)
    const int f = (int)(idx & 127);
    const float di = dis[i];
    const float v = acc[idx] + hpre[idx] * di * di + bias[f];
    acc[idx] = fmaxf(v, 0.0f);
}

// ---------------------------------------------------------------------------
// Global mean pool: atomic segment-sum + counts (one wave per node).
// ---------------------------------------------------------------------------
__global__ __launch_bounds__(256)
void pool_sum(const float* __restrict__ h, const int* __restrict__ batch,
              float* __restrict__ gsum, float* __restrict__ counts, int N) {
    const int i = blockIdx.x * 8 + (threadIdx.x >> 5);
    if (i >= N) return;
    const int lane = threadIdx.x & 31;
    const int b = batch[i];
    const float4 v = ((const float4*)(h + (size_t)i * 128))[lane];
    float* o = gsum + (size_t)b * 128 + (size_t)lane * 4;
    atomicAdd(o + 0, v.x);
    atomicAdd(o + 1, v.y);
    atomicAdd(o + 2, v.z);
    atomicAdd(o + 3, v.w);
    if (lane == 0) atomicAdd(&counts[b], 1.0f);
}

__global__ __launch_bounds__(256)
void pool_div(float* __restrict__ gsum, const float* __restrict__ counts, size_t total) {
    size_t idx = (size_t)blockIdx.x * 256 + threadIdx.x;
    if (idx >= total) return;
    gsum[idx] /= fmaxf(counts[idx >> 7], 1.0f);
}

// ---------------------------------------------------------------------------
static inline size_t alignup(size_t x) { return (x + 255) & ~(size_t)255; }

extern "C" void kernel_launch(void* const* d_in, const int* in_sizes, int n_in,
                              void* d_out, int out_size, void* d_ws, size_t ws_size,
                              hipStream_t stream) {
    const float* x     = (const float*)d_in[0];
    const int*   ei    = (const int*)d_in[1];
    const int*   batch = (const int*)d_in[2];
    const float* W1    = (const float*)d_in[3];
    const float* b1    = (const float*)d_in[4];
    const float* W2    = (const float*)d_in[5];
    const float* b2    = (const float*)d_in[6];
    const float* Wlin  = (const float*)d_in[7];
    const float* blin  = (const float*)d_in[8];

    const int N = in_sizes[2];           // 50000 nodes
    const int E = in_sizes[1] / 2;       // 1.6M edges
    const int H = in_sizes[4];           // 128
    const int D = in_sizes[0] / N;       // 128
    const int O = in_sizes[8];           // 138
    const int G = out_size / O;          // 512 graphs

    const int* src = ei;                 // edge_index[0]
    const int* dst = ei + E;             // edge_index[1]

    // Workspace layout
    char* w = (char*)d_ws;
    float* dis    = (float*)w;  w += alignup((size_t)N * sizeof(float));
    float* bufA   = (float*)w;  w += alignup((size_t)N * H * sizeof(float));
    float* bufB   = (float*)w;  w += alignup((size_t)N * H * sizeof(float));
    float* gsum   = (float*)w;  w += alignup((size_t)G * H * sizeof(float));
    float* counts = (float*)w;

    const size_t NH = (size_t)N * H;
    const size_t GH = (size_t)G * H;
    const int TPB = 256;
    auto blocks = [](size_t n, int t) { return (unsigned)((n + t - 1) / t); };

    // ---- GCN normalization: deg (init 1.0 for self-loop) -> deg^{-1/2} ----
    fill_f32<<<blocks(N, TPB), TPB, 0, stream>>>(dis, 1.0f, (size_t)N);
    degree_kernel<<<blocks(E, TPB), TPB, 0, stream>>>(dst, dis, E);
    rsqrt_kernel<<<blocks(N, TPB), TPB, 0, stream>>>(dis, N);

    // ---- Layer 1: h_pre = x @ W1  (WMMA) ----
    wmma_gemm_f32<<<dim3(N / 16, H / 16), 32, 0, stream>>>(x, W1, nullptr, bufA, D, H, 0);
    // scatter over edges, self-loop folded into finish
    fill_f32<<<blocks(NH, TPB), TPB, 0, stream>>>(bufB, 0.0f, NH);
    scatter_edges<<<blocks((size_t)E * 32, TPB), TPB, 0, stream>>>(bufA, dis, src, dst, bufB, E);
    finish_layer<<<blocks(NH, TPB), TPB, 0, stream>>>(bufB, bufA, dis, b1, NH);

    // ---- Layer 2: h_pre = h1 @ W2 (WMMA) ----
    wmma_gemm_f32<<<dim3(N / 16, H / 16), 32, 0, stream>>>(bufB, W2, nullptr, bufA, H, H, 0);
    fill_f32<<<blocks(NH, TPB), TPB, 0, stream>>>(bufB, 0.0f, NH);
    scatter_edges<<<blocks((size_t)E * 32, TPB), TPB, 0, stream>>>(bufA, dis, src, dst, bufB, E);
    finish_layer<<<blocks(NH, TPB), TPB, 0, stream>>>(bufB, bufA, dis, b2, NH);

    // ---- Global mean pool ----
    fill_f32<<<blocks(GH, TPB), TPB, 0, stream>>>(gsum, 0.0f, GH);
    fill_f32<<<blocks(G, TPB), TPB, 0, stream>>>(counts, 0.0f, (size_t)G);
    pool_sum<<<blocks((size_t)N * 32, TPB), TPB, 0, stream>>>(bufB, batch, gsum, counts, N);
    pool_div<<<blocks(GH, TPB), TPB, 0, stream>>>(gsum, counts, GH);

    // ---- Final linear: logits = g @ Wlin + blin (WMMA, O = 138 guarded) ----
    wmma_gemm_f32<<<dim3(G / 16, (O + 15) / 16), 32, 0, stream>>>(
        gsum, Wlin, blin, (float*)d_out, H, O, 0);
}